// SpikeLatencyLIFEncoder_9672266350606
// MI455X (gfx1250) — compile-verified
//
#include <hip/hip_runtime.h>

// SpikeLatencyLIFEncoder for MI455X (gfx1250).
//
// v <- 0.9*v + 0.1*I ; z = (v > 1) ; v <- z ? 0 : v ;
// spike = (mask==0) ? z : 0 ; mask += z ; repeated 100 steps.
//
// Streaming-store bound: 419 MB out vs 0.5 GFLOP. Strategy:
//  * 4 elements per thread, recurrence fully in VGPRs
//  * input staged via CDNA5 async global->LDS copy (ASYNCcnt path)
//  * one non-temporal global_store_b128 per thread per timestep

#define SEQ_LENGTH 100
#define DECAY 0.9f
#define GAIN  0.1f   // DT * TAU_MEM_INV
#define V_TH  1.0f

typedef float float4v __attribute__((ext_vector_type(4)));

__global__ __launch_bounds__(256)
void lif_latency_kernel(const float* __restrict__ in,
                        float* __restrict__ out,
                        int n /* total elements, multiple of 4 */) {
    __shared__ float4v stage[256];

    const int tid = threadIdx.x;
    const long long idx4 = ((long long)blockIdx.x * 256 + tid) * 4;
    if (idx4 >= n) return;

    // ---- CDNA5 async global -> LDS input staging ------------------------
    // LDS byte offset of this thread's slot. On AMDGPU the low 32 bits of a
    // flat shared pointer are the LDS byte offset.
    unsigned lds_off = (unsigned)(unsigned long long)&stage[tid];
    unsigned long long gaddr = (unsigned long long)(in + idx4);
    asm volatile(
        "global_load_async_to_lds_b128 %0, %1, off\n\t"
        "s_wait_asynccnt 0"
        :
        : "v"(lds_off), "v"(gaddr)
        : "memory");

    // Read the staged tile (ds_load_b128); asm above has a memory clobber so
    // this cannot be hoisted before the async copy completes.
    const float4v I = stage[tid];

    // ---- register-resident 100-step scan --------------------------------
    float a[4], v[4], mask[4];
    #pragma unroll
    for (int e = 0; e < 4; ++e) {
        a[e]    = I[e] * GAIN;   // constant drive term per step
        v[e]    = 0.0f;
        mask[e] = 0.0f;
    }

    float* outp = out + idx4;
    #pragma unroll 4
    for (int t = 0; t < SEQ_LENGTH; ++t) {
        float4v s;
        #pragma unroll
        for (int e = 0; e < 4; ++e) {
            v[e] = __builtin_fmaf(v[e], DECAY, a[e]);     // leaky integrate
            const float z = (v[e] > V_TH) ? 1.0f : 0.0f;  // heaviside
            s[e] = (mask[e] == 0.0f) ? z : 0.0f;          // latency: first only
            v[e] = (z != 0.0f) ? 0.0f : v[e];             // reset on spike
            mask[e] += z;
        }
        // Output plane t: 419 MB total stream, never re-read -> NT hint so we
        // don't thrash the 192 MB L2.
        __builtin_nontemporal_store(s, (float4v*)outp);
        outp += n;
    }
}

extern "C" void kernel_launch(void* const* d_in, const int* in_sizes, int n_in,
                              void* d_out, int out_size, void* d_ws, size_t ws_size,
                              hipStream_t stream) {
    const float* in = (const float*)d_in[0];
    float* out = (float*)d_out;
    const int n = in_sizes[0];                     // 256*4096 = 1,048,576
    const int per_block = 256 * 4;                 // 4 elements per thread
    const int blocks = (n + per_block - 1) / per_block;
    lif_latency_kernel<<<blocks, 256, 0, stream>>>(in, out, n);
}